// GNNModel_50457275793790
// MI455X (gfx1250) — compile-verified
//
#include <hip/hip_runtime.h>

typedef __attribute__((ext_vector_type(2))) float v2f;
typedef __attribute__((ext_vector_type(8))) float v8f;

#define D 128
#define TILE_M 16
#define TILES_PER_BLOCK 8    // 128 nodes per block
#define APAD 260             // LDS row stride (floats), padded vs 256 for bank spread

// -------------------------------------------------------------------------
// zero a float buffer
__global__ void zero_kernel(float* __restrict__ p, long long n) {
    long long i = (long long)blockIdx.x * 256 + threadIdx.x;
    if (i < n) p[i] = 0.0f;
}

// -------------------------------------------------------------------------
// edge scatter: sum[dst,:] += x[src,:]  (+ optional degree accumulation)
// one thread per (edge, 4-float chunk): E*32 threads
__global__ void scatter_kernel(const float* __restrict__ x,
                               const int* __restrict__ ei,   // [2,E] int32
                               float* __restrict__ sum,
                               float* __restrict__ deg,      // may be null
                               int n_edges) {
    long long tid = (long long)blockIdx.x * 256 + threadIdx.x;
    long long total = (long long)n_edges * 32;
    if (tid >= total) return;
    int e  = (int)(tid >> 5);
    int c4 = (int)(tid & 31) * 4;
    int src = ei[e];
    int dst = ei[n_edges + e];
    const float4 v = *(const float4*)(x + (size_t)src * D + c4);
    float* sp = sum + (size_t)dst * D + c4;
    unsafeAtomicAdd(sp + 0, v.x);
    unsafeAtomicAdd(sp + 1, v.y);
    unsafeAtomicAdd(sp + 2, v.z);
    unsafeAtomicAdd(sp + 3, v.w);
    if (deg != nullptr && c4 == 0) unsafeAtomicAdd(deg + dst, 1.0f);
}

// -------------------------------------------------------------------------
// out[i,:] = act( (sum[i,:]/max(deg[i],1)) @ Wl + bias + xin[i,:] @ Wr )
// Fused as single K=256 GEMM:  [agg | xin] @ [Wl ; Wr]
// Block: 256 threads = 8 waves. Wave w owns output columns [16w,16w+16).
// B fragments (the wave's 16-column slab of [Wl;Wr]) preloaded to registers,
// then 8 node tiles of 16 rows are swept, A' staged in LDS per tile.
__global__ __launch_bounds__(256)
void sage_linear_kernel(const float* __restrict__ sum,
                        const float* __restrict__ deg,
                        const float* __restrict__ xin,
                        const float* __restrict__ Wl,
                        const float* __restrict__ Wr,
                        const float* __restrict__ bias,
                        float* __restrict__ out,
                        int n_nodes, int do_relu) {
    __shared__ float As[TILE_M * APAD];

    const int t     = threadIdx.x;
    const int lane  = t & 31;
    const int wave  = t >> 5;        // 0..7
    const int n0    = wave * 16;     // output column base
    const int rm    = lane & 15;     // A row / C column within tile
    const int khalf = lane >> 4;     // 0 or 1 (selects K pair within step)

    // ---- preload B fragments: W' = [Wl ; Wr], rows k = kk*4 + 2*khalf ----
    v2f bfrag[64];
#pragma unroll
    for (int kk = 0; kk < 64; ++kk) {
        int k = kk * 4 + 2 * khalf;                 // pair (k, k+1) never crosses 128
        const float* W = (k < 128) ? Wl : Wr;
        int kr = (k < 128) ? k : (k - 128);
        bfrag[kk].x = W[kr * D + n0 + rm];
        bfrag[kk].y = W[(kr + 1) * D + n0 + rm];
    }
    const float bv = bias[n0 + rm];

    const int tile_base0 = blockIdx.x * (TILES_PER_BLOCK * TILE_M);

    for (int tt = 0; tt < TILES_PER_BLOCK; ++tt) {
        int tile_base = tile_base0 + tt * TILE_M;
        if (tile_base >= n_nodes) break;            // uniform across block

        __syncthreads();
        // ---- stage A' = [agg | xin] tile: 16 rows x 256 cols ----
        for (int i = t; i < TILE_M * 256; i += 256) {
            int row  = i >> 8;
            int c    = i & 255;
            int node = tile_base + row;
            float v = 0.0f;
            if (node < n_nodes) {
                if (c < D) {
                    float rd = 1.0f / fmaxf(deg[node], 1.0f);
                    v = sum[(size_t)node * D + c] * rd;
                } else {
                    v = xin[(size_t)node * D + (c - D)];
                }
            }
            As[row * APAD + c] = v;
        }
        __syncthreads();

        // ---- K=256 accumulation with V_WMMA_F32_16X16X4_F32 ----
        v8f acc = {};
#pragma unroll
        for (int kk = 0; kk < 64; ++kk) {
            int koff = kk * 4 + 2 * khalf;
            v2f a;
            a.x = As[rm * APAD + koff];
            a.y = As[rm * APAD + koff + 1];
            acc = __builtin_amdgcn_wmma_f32_16x16x4_f32(
                false, a, false, bfrag[kk], (short)0, acc, false, false);
        }

        // ---- epilogue: bias, optional ReLU, store C tile ----
#pragma unroll
        for (int j = 0; j < 8; ++j) {
            float v = acc[j] + bv;
            if (do_relu) v = fmaxf(v, 0.0f);
            int m    = j + khalf * 8;               // C row within tile
            int node = tile_base + m;
            if (node < n_nodes)
                out[(size_t)node * D + n0 + rm] = v;
        }
    }
}

// -------------------------------------------------------------------------
extern "C" void kernel_launch(void* const* d_in, const int* in_sizes, int n_in,
                              void* d_out, int out_size, void* d_ws, size_t ws_size,
                              hipStream_t stream) {
    (void)n_in; (void)out_size; (void)ws_size;

    const float* x   = (const float*)d_in[0];
    const int*   ei  = (const int*)d_in[1];
    const float* W1l = (const float*)d_in[2];
    const float* b1l = (const float*)d_in[3];
    const float* W1r = (const float*)d_in[4];
    const float* W2l = (const float*)d_in[5];
    const float* b2l = (const float*)d_in[6];
    const float* W2r = (const float*)d_in[7];
    float* out = (float*)d_out;

    const int n_nodes = in_sizes[0] / D;     // 50000
    const int n_edges = in_sizes[1] / 2;     // 800000

    // workspace layout (floats): [deg | sum | h]
    float* ws  = (float*)d_ws;
    int degN   = ((n_nodes + 15) / 16) * 16;
    float* deg = ws;
    float* sum = ws + degN;
    float* h   = sum + (size_t)n_nodes * D;

    const long long zc1 = (long long)degN + (long long)n_nodes * D;
    zero_kernel<<<(int)((zc1 + 255) / 256), 256, 0, stream>>>(deg, zc1);

    const long long sthreads = (long long)n_edges * 32;
    const int sblocks = (int)((sthreads + 255) / 256);
    scatter_kernel<<<sblocks, 256, 0, stream>>>(x, ei, sum, deg, n_edges);

    const int lblocks = (n_nodes + TILES_PER_BLOCK * TILE_M - 1) / (TILES_PER_BLOCK * TILE_M);
    sage_linear_kernel<<<lblocks, 256, 0, stream>>>(sum, deg, x, W1l, W1r, b1l, h,
                                                    n_nodes, /*relu=*/1);

    const long long zc2 = (long long)n_nodes * D;
    zero_kernel<<<(int)((zc2 + 255) / 256), 256, 0, stream>>>(sum, zc2);

    scatter_kernel<<<sblocks, 256, 0, stream>>>(h, ei, sum, nullptr, n_edges);

    sage_linear_kernel<<<lblocks, 256, 0, stream>>>(sum, deg, h, W2l, W2r, b2l, out,
                                                    n_nodes, /*relu=*/0);
}